// SAE_41042707481043
// MI455X (gfx1250) — compile-verified
//
#include <hip/hip_runtime.h>
#include <hip/hip_bf16.h>

typedef __attribute__((ext_vector_type(2))) float v2f;
typedef __attribute__((ext_vector_type(8))) float v8f;
typedef __attribute__((ext_vector_type(4))) int   v4i;

#define BROWS 4096
#define DIN   768
#define DSAE  24576
#define TOPK  32
#define AUXK  256
#define DEADTH 1000
#define EPS   1e-5f
#define NCHUNK (DIN / 16)   // 48 K-chunks of 16

#if defined(__gfx1250__) && __has_builtin(__builtin_amdgcn_global_load_async_to_lds_b128)
#define HAVE_ASYNC 1
#if __has_builtin(__builtin_amdgcn_s_wait_asynccnt)
#define WAIT_ASYNC(n) __builtin_amdgcn_s_wait_asynccnt(n)
#else
#define WAIT_ASYNC(n) asm volatile("s_wait_asynccnt " #n ::: "memory")
#endif
typedef __attribute__((address_space(1))) v4i gv4i;   // global int4
typedef __attribute__((address_space(3))) v4i lv4i;   // LDS int4
#endif

// ---------------- LayerNorm (ddof=1) + fold (-b_dec) ----------------
__launch_bounds__(256)
__global__ void ln_kernel(const float* __restrict__ x, const float* __restrict__ b_dec,
                          float* __restrict__ xn, float* __restrict__ mu, float* __restrict__ sd) {
    const int r = blockIdx.x;
    const int tid = threadIdx.x;
    const float* xr = x + (size_t)r * DIN;
    __shared__ float red[256];

    float s = 0.f;
    for (int i = tid; i < DIN; i += 256) s += xr[i];
    red[tid] = s; __syncthreads();
    for (int off = 128; off > 0; off >>= 1) { if (tid < off) red[tid] += red[tid + off]; __syncthreads(); }
    const float m = red[0] / (float)DIN;
    __syncthreads();

    float v = 0.f;
    for (int i = tid; i < DIN; i += 256) { float d = xr[i] - m; v += d * d; }
    red[tid] = v; __syncthreads();
    for (int off = 128; off > 0; off >>= 1) { if (tid < off) red[tid] += red[tid + off]; __syncthreads(); }
    const float stdv = sqrtf(red[0] / (float)(DIN - 1));   // unbiased, like torch std
    if (tid == 0) { mu[r] = m; sd[r] = stdv; }
    const float inv = 1.f / (stdv + EPS);
    for (int i = tid; i < DIN; i += 256)
        xn[(size_t)r * DIN + i] = (xr[i] - m) * inv - b_dec[i];
}

// ---------------- dead-neuron byte mask ----------------
__launch_bounds__(256)
__global__ void dead_kernel(const long long* __restrict__ stats, unsigned char* __restrict__ dead) {
    int i = blockIdx.x * 256 + threadIdx.x;
    if (i < DSAE) dead[i] = (stats[i] > (long long)DEADTH) ? 1 : 0;
}

// ---------------- fp32 WMMA GEMM: pre = A @ W_enc + b_enc ----------------
// Block tile 256x128, 8 waves, each wave: 32 rows x 128 cols (2x8 v8f accumulators).
// Each B fragment feeds 2 WMMAs; double-buffered LDS tiles fed by async copies.
__launch_bounds__(256)
__global__ void gemm_kernel(const float* __restrict__ A, const float* __restrict__ W,
                            const float* __restrict__ b_enc, float* __restrict__ pre) {
    __shared__ float As[2][256][20];   // row = 80B: 16B-aligned b128 chunks, even-k 8B-aligned frags
    __shared__ float Bs[2][16][128];

    const int tid  = threadIdx.x;
    const int wave = tid >> 5;
    const int lane = tid & 31;
    const int l16  = lane & 15;
    const int khalf = (lane >> 4) * 2;   // fp32 frag: lanes 0-15 hold K 0/1, lanes 16-31 hold K 2/3
    const int n0 = blockIdx.x * 128;
    const int m0 = blockIdx.y * 256;

    v8f acc[2][8];
#pragma unroll
    for (int nt = 0; nt < 8; ++nt) {
        // C/D layout: all 8 VGPRs of a lane share N = nt*16 + l16
        float b = b_enc[n0 + nt * 16 + l16];
        acc[0][nt] = (v8f){b, b, b, b, b, b, b, b};
        acc[1][nt] = (v8f){b, b, b, b, b, b, b, b};
    }

    auto computeTile = [&](int bsel) {
#pragma unroll
        for (int k4 = 0; k4 < 16; k4 += 4) {
            v2f a0, a1;
            a0.x = As[bsel][wave * 32 + l16][k4 + khalf];
            a0.y = As[bsel][wave * 32 + l16][k4 + khalf + 1];
            a1.x = As[bsel][wave * 32 + 16 + l16][k4 + khalf];
            a1.y = As[bsel][wave * 32 + 16 + l16][k4 + khalf + 1];
#pragma unroll
            for (int nt = 0; nt < 8; ++nt) {
                v2f bf;
                bf.x = Bs[bsel][k4 + khalf][nt * 16 + l16];
                bf.y = Bs[bsel][k4 + khalf + 1][nt * 16 + l16];
                acc[0][nt] = __builtin_amdgcn_wmma_f32_16x16x4_f32(
                    false, a0, false, bf, (short)0, acc[0][nt], false, false);
                acc[1][nt] = __builtin_amdgcn_wmma_f32_16x16x4_f32(
                    false, a1, false, bf, (short)0, acc[1][nt], false, false);
            }
        }
    };

#if HAVE_ASYNC
    // 6 async b128 copies per thread per tile (4 for A, 2 for B)
    auto issueTile = [&](int kt, int bsel) {
        for (int i = tid; i < 1024; i += 256) {          // A: 256x16, 16B chunks
            int m = i >> 2, q = (i & 3) * 4;
            __builtin_amdgcn_global_load_async_to_lds_b128(
                (gv4i*)(A + (size_t)(m0 + m) * DIN + kt + q),
                (lv4i*)&As[bsel][m][q], 0, 0);
        }
        for (int i = tid; i < 512; i += 256) {           // B: 16x128, 16B chunks
            int kk = i >> 5, q = (i & 31) * 4;
            __builtin_amdgcn_global_load_async_to_lds_b128(
                (gv4i*)(W + (size_t)(kt + kk) * DSAE + n0 + q),
                (lv4i*)&Bs[bsel][kk][q], 0, 0);
        }
    };

    int buf = 0;
    issueTile(0, 0);
    for (int c = 0; c < NCHUNK - 1; ++c) {
        issueTile((c + 1) * 16, buf ^ 1);   // prefetch next tile into other buffer
        WAIT_ASYNC(6);                      // tile c complete; tile c+1 (6 copies) still in flight
        __syncthreads();
        computeTile(buf);
        __syncthreads();                    // buf^1 safe to refill next iteration
        buf ^= 1;
    }
    WAIT_ASYNC(0);
    __syncthreads();
    computeTile(buf);
#else
    for (int kt = 0; kt < DIN; kt += 16) {
        for (int i = tid; i < 1024; i += 256) {
            int m = i >> 2, q = (i & 3) * 4;
            float4 vv = *(const float4*)&A[(size_t)(m0 + m) * DIN + kt + q];
            As[0][m][q + 0] = vv.x; As[0][m][q + 1] = vv.y; As[0][m][q + 2] = vv.z; As[0][m][q + 3] = vv.w;
        }
        for (int i = tid; i < 512; i += 256) {
            int kk = i >> 5, q = (i & 31) * 4;
            float4 vv = *(const float4*)&W[(size_t)(kt + kk) * DSAE + n0 + q];
            Bs[0][kk][q + 0] = vv.x; Bs[0][kk][q + 1] = vv.y; Bs[0][kk][q + 2] = vv.z; Bs[0][kk][q + 3] = vv.w;
        }
        __syncthreads();
        computeTile(0);
        __syncthreads();
    }
#endif

    // Store: VGPR v -> row (+v) for lanes 0-15, (+8) for lanes 16-31, per sub-tile
#pragma unroll
    for (int sub = 0; sub < 2; ++sub) {
        const int mrow = m0 + wave * 32 + sub * 16 + (lane >> 4) * 8;
#pragma unroll
        for (int nt = 0; nt < 8; ++nt) {
            const int n = n0 + nt * 16 + l16;
#pragma unroll
            for (int vg = 0; vg < 8; ++vg)
                pre[(size_t)(mrow + vg) * DSAE + n] = acc[sub][nt][vg];
        }
    }
}

// ---------------- top-k select + sparse decode, one workgroup per row ----------------
__device__ __forceinline__ unsigned fkey(float f) {
    unsigned u = __float_as_uint(f);
    return (u & 0x80000000u) ? ~u : (u | 0x80000000u);   // monotone: larger float -> larger key
}
__device__ __forceinline__ float keyval(unsigned k) {
    unsigned u = (k & 0x80000000u) ? (k & 0x7fffffffu) : ~k;
    return __uint_as_float(u);
}

__launch_bounds__(256)
__global__ void select_decode_kernel(const float* __restrict__ pre, const float* __restrict__ Wd,
                                     const float* __restrict__ b_dec, const unsigned char* __restrict__ dead,
                                     const float* __restrict__ mu, const float* __restrict__ sd,
                                     float* __restrict__ out) {
    extern __shared__ unsigned smem[];                 // keys[DSAE] (96KB) + mask[DSAE] (24KB): fits 320KB WGP LDS
    unsigned* keys = smem;
    unsigned char* mk = (unsigned char*)(keys + DSAE);

    __shared__ unsigned hist[256];
    __shared__ unsigned s_prefix, s_rem;
    __shared__ int  scnt;
    __shared__ int   selIdx[AUXK];
    __shared__ float selVal[AUXK];
    __shared__ float acc[DIN];

    const int r = blockIdx.x;
    const int tid = threadIdx.x;
    const unsigned ZKEY = 0x80000000u;                 // key(+0.0f)

    for (int i = tid; i < DSAE; i += 256) {
        keys[i] = fkey(pre[(size_t)r * DSAE + i]);
        mk[i]   = dead[i];
    }
    __syncthreads();

    // 8-bit radix select: threshold = kk-th largest key (4 histogram passes per mode)
    unsigned thr[2];
    for (int mode = 0; mode < 2; ++mode) {
        const unsigned kk = mode ? AUXK : TOPK;
        if (tid == 0) { s_rem = kk; s_prefix = 0; }
        __syncthreads();
        for (int shift = 24; shift >= 0; shift -= 8) {
            hist[tid] = 0;
            __syncthreads();
            const unsigned prefix = s_prefix;
            const unsigned mask = (shift == 24) ? 0u : (0xFFFFFFFFu << (shift + 8));
            for (int i = tid; i < DSAE; i += 256) {
                unsigned kv = mode ? (mk[i] ? keys[i] : ZKEY) : keys[i];
                if ((kv & mask) == prefix)
                    atomicAdd(&hist[(kv >> shift) & 255u], 1u);
            }
            __syncthreads();
            if (tid == 0) {
                unsigned rem = s_rem;
                int b = 255;
                for (; b > 0; --b) {
                    if (rem > hist[b]) rem -= hist[b];
                    else break;
                }
                s_prefix = prefix | ((unsigned)b << shift);
                s_rem = rem;
            }
            __syncthreads();
        }
        thr[mode] = s_prefix;
        __syncthreads();
    }

    const float m_ = mu[r], s_ = sd[r];
    for (int mode = 0; mode < 2; ++mode) {
        const int kk = mode ? AUXK : TOPK;
        const unsigned t = thr[mode];
        if (tid == 0) scnt = 0;
        for (int i = tid; i < DIN; i += 256) acc[i] = 0.f;
        __syncthreads();

        // compact selected (idx, relu(val)) pairs
        for (int i = tid; i < DSAE; i += 256) {
            unsigned kv = mode ? (mk[i] ? keys[i] : ZKEY) : keys[i];
            if (kv >= t) {
                int p = atomicAdd(&scnt, 1);
                if (p < kk) {
                    selIdx[p] = i;
                    float v = (mode && !mk[i]) ? 0.f : keyval(keys[i]);
                    selVal[p] = fmaxf(v, 0.f);
                }
            }
        }
        __syncthreads();

        // sparse decode: acc += val * W_dec[idx]; per-thread-owned acc slots (no atomics)
        const int ns = scnt < kk ? scnt : kk;
        for (int s = 0; s < ns; ++s) {
            const float v = selVal[s];
            const float* wrow = Wd + (size_t)selIdx[s] * DIN;
            for (int i = tid; i < DIN; i += 256) acc[i] += v * wrow[i];
        }
        __syncthreads();

        float* o = out + (size_t)mode * BROWS * DIN + (size_t)r * DIN;
        for (int i = tid; i < DIN; i += 256) o[i] = (acc[i] + b_dec[i]) * s_ + m_;
        __syncthreads();
    }
}

extern "C" void kernel_launch(void* const* d_in, const int* in_sizes, int n_in,
                              void* d_out, int out_size, void* d_ws, size_t ws_size,
                              hipStream_t stream) {
    const float* x      = (const float*)d_in[0];
    const float* W_enc  = (const float*)d_in[1];
    const float* W_dec  = (const float*)d_in[2];
    const float* b_enc  = (const float*)d_in[3];
    const float* b_dec  = (const float*)d_in[4];
    const long long* stats = (const long long*)d_in[5];
    float* out = (float*)d_out;

    // Workspace layout: pre (402MB) | xn (12.6MB) | mu | sd | dead mask
    float* pre = (float*)d_ws;
    float* xn  = pre + (size_t)BROWS * DSAE;
    float* mu  = xn + (size_t)BROWS * DIN;
    float* sd  = mu + BROWS;
    unsigned char* dead = (unsigned char*)(sd + BROWS);

    ln_kernel<<<BROWS, 256, 0, stream>>>(x, b_dec, xn, mu, sd);
    dead_kernel<<<(DSAE + 255) / 256, 256, 0, stream>>>(stats, dead);

    dim3 gg(DSAE / 128, BROWS / 256);
    gemm_kernel<<<gg, 256, 0, stream>>>(xn, W_enc, b_enc, pre);

    const size_t dyn = (size_t)DSAE * sizeof(unsigned) + (size_t)DSAE;  // 120 KB
    (void)hipFuncSetAttribute((const void*)select_decode_kernel,
                              hipFuncAttributeMaxDynamicSharedMemorySize, (int)dyn);
    select_decode_kernel<<<BROWS, 256, dyn, stream>>>(pre, W_dec, b_dec, dead, mu, sd, out);
}